// PlateCrossEntropyLoss_46901042872883
// MI455X (gfx1250) — compile-verified
//
#include <hip/hip_runtime.h>
#include <math.h>

#define N_CLASSES   1108
#define GROUP_SIZE  277
#define WAVES_PER_BLOCK 8
#define BLOCK_THREADS   256

typedef __attribute__((ext_vector_type(2))) float v2f;
typedef __attribute__((ext_vector_type(8))) float v8f;

// ---------- CDNA5 async global->LDS staging (guarded; falls back to plain loads) ----------
__device__ __forceinline__ void async_copy_dword(const int* gsrc, int* lds_dst) {
#if defined(__gfx1250__) && __has_builtin(__builtin_amdgcn_global_load_async_to_lds_b32)
    __builtin_amdgcn_global_load_async_to_lds_b32(
        (__attribute__((address_space(1))) int*)((int*)gsrc),
        (__attribute__((address_space(3))) int*)(lds_dst),
        0, 0);
#else
    *lds_dst = *gsrc;
#endif
}

__device__ __forceinline__ void async_wait_zero() {
#if defined(__gfx1250__) && __has_builtin(__builtin_amdgcn_global_load_async_to_lds_b32)
#if __has_builtin(__builtin_amdgcn_s_wait_asynccnt)
    __builtin_amdgcn_s_wait_asynccnt(0);
#else
    asm volatile("s_wait_asynccnt 0" ::: "memory");
#endif
#endif
}

// ---------- Kernel 1: one wave32 per sample; per-block partial NLL sums ----------
__global__ __launch_bounds__(BLOCK_THREADS)
void plate_ce_partial(const float* __restrict__ output,
                      const int*   __restrict__ target,
                      const int*   __restrict__ masks,      // [4][277]
                      const int*   __restrict__ group_of,   // [1108]
                      float*       __restrict__ partial) {  // [gridDim.x]
    __shared__ int   lds_masks[N_CLASSES];   // 4*277 = 1108 ints
    __shared__ float wave_nll[WAVES_PER_BLOCK];

    const int tid = threadIdx.x;
    for (int i = tid; i < N_CLASSES; i += BLOCK_THREADS)
        async_copy_dword(masks + i, &lds_masks[i]);
    async_wait_zero();
    __syncthreads();

    const int lane   = tid & 31;
    const int wid    = tid >> 5;
    const int sample = blockIdx.x * WAVES_PER_BLOCK + wid;

    const int    t     = target[sample];
    const int    gid   = group_of[t];
    const size_t row   = (size_t)sample * N_CLASSES;
    const int    mbase = gid * GROUP_SIZE;

    // Gather the group's 277 logits: 9 strided passes per lane, coalesced-ish
    // within the 4.4KB row (every cacheline of the row is touched anyway).
    float vals[9];
    float m = -3.0e38f;
#pragma unroll
    for (int it = 0; it < 9; ++it) {
        const int i = it * 32 + lane;
        float x = -3.0e38f;
        if (i < GROUP_SIZE) {
            const int idx = lds_masks[mbase + i];
            x = output[row + idx];
        }
        vals[it] = x;
        m = fmaxf(m, x);
    }
#pragma unroll
    for (int off = 16; off > 0; off >>= 1)
        m = fmaxf(m, __shfl_xor(m, off, 32));

    float s = 0.0f;
#pragma unroll
    for (int it = 0; it < 9; ++it)
        s += __expf(vals[it] - m);        // invalid lanes: exp(-3e38 - m) -> 0
#pragma unroll
    for (int off = 16; off > 0; off >>= 1)
        s += __shfl_xor(s, off, 32);

    if (lane == 0) {
        const float xt = output[row + t];             // masks[gid][pos] == t
        wave_nll[wid] = -(xt - m - __logf(s));
    }
    __syncthreads();

    if (tid == 0) {
        float acc = 0.0f;
#pragma unroll
        for (int w = 0; w < WAVES_PER_BLOCK; ++w) acc += wave_nll[w];
        partial[blockIdx.x] = acc;                    // fixed order: deterministic
    }
}

// ---------- Kernel 2: single-wave deterministic final reduce (WMMA f32 path) ----------
__global__ __launch_bounds__(32)
void plate_ce_reduce(const float* __restrict__ partial, float* __restrict__ out,
                     int n, float scale) {
    const int lane = threadIdx.x;   // exactly 32 threads, EXEC all-ones
#if defined(__gfx1250__) && __has_builtin(__builtin_amdgcn_wmma_f32_16x16x4_f32)
    // D = A x B + C with B == ones(4x16): every WMMA folds 64 f32 values into
    // the f32 accumulator's row sums; column N=0 lives in lanes 0 (M=0..7) and
    // 16 (M=8..15), one row per acc VGPR.
    v8f acc = {};
    v2f ones; ones.x = 1.0f; ones.y = 1.0f;
    for (int base = 0; base < n; base += 64) {      // n == 8192, multiple of 64
        v2f a;
        a.x = partial[base + lane];
        a.y = partial[base + 32 + lane];
        acc = __builtin_amdgcn_wmma_f32_16x16x4_f32(
            /*neg_a=*/false, a, /*neg_b=*/false, ones,
            /*c_mod=*/(short)0, acc, /*reuse_a=*/false, /*reuse_b=*/false);
    }
    float s = acc[0] + acc[1] + acc[2] + acc[3] + acc[4] + acc[5] + acc[6] + acc[7];
    const float total = __shfl(s, 0, 32) + __shfl(s, 16, 32);
    if (lane == 0) out[0] = total * scale;
#else
    float s = 0.0f;
    for (int i = lane; i < n; i += 32) s += partial[i];
#pragma unroll
    for (int off = 16; off > 0; off >>= 1) s += __shfl_xor(s, off, 32);
    if (lane == 0) out[0] = s * scale;
#endif
}

extern "C" void kernel_launch(void* const* d_in, const int* in_sizes, int n_in,
                              void* d_out, int out_size, void* d_ws, size_t ws_size,
                              hipStream_t stream) {
    const float* output   = (const float*)d_in[0];
    const int*   target   = (const int*)d_in[1];
    const int*   masks    = (const int*)d_in[2];
    const int*   group_of = (const int*)d_in[3];
    // d_in[4] (pos_of) unused: x_target is read directly at column `target`.

    const int batch   = in_sizes[1];                 // 65536
    const int nblocks = batch / WAVES_PER_BLOCK;     // 8192

    float* partial = (float*)d_ws;                   // 8192 f32 = 32 KB scratch

    plate_ce_partial<<<nblocks, BLOCK_THREADS, 0, stream>>>(
        output, target, masks, group_of, partial);
    plate_ce_reduce<<<1, 32, 0, stream>>>(
        partial, (float*)d_out, nblocks, 1.0f / (float)batch);
}